// FullTensorProductSparse_747324309793
// MI455X (gfx1250) — compile-verified
//
#include <hip/hip_runtime.h>
#include <complex>
#include <vector>
#include <numeric>
#include <algorithm>
#include <cmath>
#include <cstring>

// ---------------------------------------------------------------------------
// Full e3nn tensor product 64x0e+32x1o+16x2e (x) itself, N=1024 rows.
// Per row: 240 in / 240 in -> 57600 out f32.  Store-bound (236 MB out).
//  * Input rows staged into LDS via TENSOR_LOAD_TO_LDS (TDM, TENSORcnt).
//  * l3=0 paths ((0,0)->0, (1,1)->0, (2,2)->0) on v_wmma_f32_16x16x4_f32.
//  * Remaining 16 paths: sparse CG contractions on VALU from LDS.
// ---------------------------------------------------------------------------

#define IN_DIM   240
#define OUT_DIM  57600
#define NVALU    16
#define MAXNZ    2048
#define MAXK     160
#define LDSNZ    1280

struct DevTables {
  // per VALU path: [m1mask, m1shift, d1, d2, d3, a1, a2, out_off, kstart, pairs]
  int   pd[NVALU][10];
  int   kstarts[MAXK];
  int   nz_ij[MAXNZ];   // i | (j<<8)
  float nz_v[MAXNZ];
  float s000, s110, s220;  // diagonal CG scales of the three l3=0 WMMA paths
  int   total_nz;
};

typedef __attribute__((ext_vector_type(2))) float v2f;
typedef __attribute__((ext_vector_type(8))) float v8f;
typedef __attribute__((ext_vector_type(4))) unsigned int v4u;
typedef __attribute__((ext_vector_type(8))) int v8i;
typedef __attribute__((ext_vector_type(4))) int v4i;

// 32-bit LDS byte offset of a __shared__ object (generic -> AS(3) -> int).
__device__ static inline unsigned lds_off_u32(const void* p) {
  return (unsigned)(unsigned long long)
      (__attribute__((address_space(3))) const char*)p;
}

// Issue one TDM descriptor: contiguous row of 240 f32, global -> LDS.
// D# layout per CDNA5 ISA 8.3/8.4 (group0: count/lds/global/type,
// group1: data_size=4B, tensor_dim0=tile_dim0=240, dim1=1, strides=240).
__device__ static inline void tdm_load_row(const float* gsrc, unsigned lds_byte) {
  unsigned long long ga = (unsigned long long)gsrc;
  v4u g0;
  g0[0] = 1u;                                         // count=1, user mode
  g0[1] = lds_byte;                                   // lds_addr
  g0[2] = (unsigned)(ga & 0xffffffffu);               // global_addr[31:0]
  g0[3] = (unsigned)((ga >> 32) & 0x01ffffffu)        // global_addr[56:32]
        | (2u << 30);                                 // type=2 ("image")
  v8i g1;
  g1[0] = (int)(2u << 16);          // workgroup_mask=0, data_size=2 (4 bytes)
  g1[1] = (int)(240u << 16);        // tensor_dim0[15:0] (bits 63:48)
  g1[2] = (int)(1u << 16);          // tensor_dim0 hi=0 | tensor_dim1 lo=1
  g1[3] = (int)(240u << 16);        // tensor_dim1 hi=0 | tile_dim0=240
  g1[4] = 1;                        // tile_dim1=1, tile_dim2=0
  g1[5] = 240;                      // tensor_dim0_stride[31:0]
  g1[6] = (int)(240u << 16);        // stride0 hi=0 | tensor_dim1_stride lo=240
  g1[7] = 0;                        // tensor_dim1_stride hi
  v4i z4 = {0, 0, 0, 0};
#if __clang_major__ >= 23
  v8i z8 = {0, 0, 0, 0, 0, 0, 0, 0};
  __builtin_amdgcn_tensor_load_to_lds(g0, g1, z4, z4, z8, 0);
#else
  __builtin_amdgcn_tensor_load_to_lds(g0, g1, z4, z4, 0);
#endif
}

__global__ __launch_bounds__(256) void ftp_kernel(
    const float* __restrict__ in1, const float* __restrict__ in2,
    float* __restrict__ out, const DevTables* __restrict__ tb, int total_nz)
{
  __shared__ float x1s[IN_DIM];
  __shared__ float x2s[IN_DIM];
  __shared__ int   s_ij[LDSNZ];
  __shared__ float s_v[LDSNZ];
  __shared__ int   s_kst[MAXK];

  const int tid = threadIdx.x;
  const int n   = blockIdx.x;
  const int wave = tid >> 5;
  const int lane = tid & 31;

  // Wave 0 drives the Tensor Data Mover for both input rows.
  if (wave == 0) {
    tdm_load_row(in1 + (size_t)n * IN_DIM, lds_off_u32(x1s));
    tdm_load_row(in2 + (size_t)n * IN_DIM, lds_off_u32(x2s));
  }
  // Meanwhile every thread helps stage the CG tables.
  for (int i = tid; i < total_nz; i += 256) {
    s_ij[i] = tb->nz_ij[i];
    s_v[i]  = tb->nz_v[i];
  }
  if (tid < MAXK) s_kst[tid] = tb->kstarts[tid];
  if (wave == 0) __builtin_amdgcn_s_wait_tensorcnt(0);
  __syncthreads();

  float* __restrict__ orow = out + (size_t)n * OUT_DIM;

  // ------------------- WMMA section: 21 16x16 tiles over 8 waves ----------
  const bool lo  = lane < 16;
  const int  l16 = lane & 15;
  const float s000 = tb->s000, s110 = tb->s110, s220 = tb->s220;

  for (int job = wave; job < 21; job += 8) {   // wave-uniform: EXEC stays full
    v8f c = {0.f,0.f,0.f,0.f,0.f,0.f,0.f,0.f};
    v2f a, b;
    float* obase;
    int v0, u0, mul;
    if (job < 16) {
      // path (0,0)->0 : c[v,u] = s000 * x2[v] * x1[u]   (rank-1, K=1)
      v0 = (job >> 2) * 16; u0 = (job & 3) * 16; mul = 64; obase = orow + 0;
      float av = x2s[v0 + l16];
      float bv = x1s[u0 + l16];
      a.x = lo ? s000 * av : 0.f;  a.y = 0.f;   // lanes>=16 carry K=2/3 -> 0
      b.x = lo ? bv : 0.f;         b.y = 0.f;
      c = __builtin_amdgcn_wmma_f32_16x16x4_f32(false, a, false, b, (short)0, c,
                                                false, false);
    } else if (job < 20) {
      // path (1,1)->0 : c[v,u] = s110 * sum_i x2[v,i]*x1[u,i]   (K=3)
      int j = job - 16;
      v0 = (j >> 1) * 16; u0 = (j & 1) * 16; mul = 32; obase = orow + 4096;
      const float* A = x2s + 64 + (v0 + l16) * 3;
      const float* B = x1s + 64 + (u0 + l16) * 3;
      a.x = s110 * A[lo ? 0 : 2];               // address-select, no branch
      a.y = lo ? s110 * A[1] : 0.f;             // K=3 slot zero
      b.x = B[lo ? 0 : 2];
      b.y = lo ? B[1] : 0.f;
      c = __builtin_amdgcn_wmma_f32_16x16x4_f32(false, a, false, b, (short)0, c,
                                                false, false);
    } else {
      // path (2,2)->0 : K=5 -> chained WMMAs (K=0..3, then K=4 through C)
      v0 = 0; u0 = 0; mul = 16; obase = orow + 5120;
      const float* A = x2s + 160 + l16 * 5;
      const float* B = x1s + 160 + l16 * 5;
      a.x = s220 * A[lo ? 0 : 2];
      a.y = s220 * A[lo ? 1 : 3];
      b.x = B[lo ? 0 : 2];
      b.y = B[lo ? 1 : 3];
      c = __builtin_amdgcn_wmma_f32_16x16x4_f32(false, a, false, b, (short)0, c,
                                                false, false);
      v2f a2, b2;
      a2.x = lo ? s220 * A[4] : 0.f;  a2.y = 0.f;
      b2.x = lo ? B[4] : 0.f;         b2.y = 0.f;
      c = __builtin_amdgcn_wmma_f32_16x16x4_f32(false, a2, false, b2, (short)0, c,
                                                false, false);
    }
    // D layout: lanes 0-15 -> M=r, N=lane; lanes 16-31 -> M=8+r, N=lane-16
    const int ncol = u0 + l16;
    const int mrow = v0 + (lo ? 0 : 8);
    #pragma unroll
    for (int r = 0; r < 8; ++r)
      obase[(mrow + r) * mul + ncol] = c[r];
  }

  // ------------------- VALU section: 16 sparse-CG paths -------------------
  for (int p = 0; p < NVALU; ++p) {
    const int* pdv = tb->pd[p];
    const int m1mask = pdv[0], m1sh = pdv[1], d1 = pdv[2], d2 = pdv[3], d3 = pdv[4];
    const int a1 = pdv[5], a2 = pdv[6], ooff = pdv[7], kst = pdv[8], pairs = pdv[9];
    float* __restrict__ op0 = orow + ooff;
    for (int t = tid; t < pairs; t += 256) {
      const int u = t & m1mask;
      const int v = t >> m1sh;
      const float* xu = x1s + a1 + u * d1;
      const float* xv = x2s + a2 + v * d2;
      float* op = op0 + t * d3;               // t == v*mul1 + u
      for (int k = 0; k < d3; ++k) {
        const int s = s_kst[kst + k];
        const int e = s_kst[kst + k + 1];
        float acc = 0.f;
        for (int q = s; q < e; ++q) {
          const int ij = s_ij[q];
          acc = fmaf(s_v[q] * xu[ij & 255], xv[ij >> 8], acc);
        }
        op[k] = acc;
      }
    }
  }
}

// ===========================================================================
// Host side: exact e3nn CG tables (same algorithm as the reference).
// ===========================================================================
static double factd(int n) { double r = 1.0; for (int i = 2; i <= n; ++i) r *= i; return r; }

static double su2_cg(int j1, int j2, int j3, int m1, int m2, int m3) {
  if (m3 != m1 + m2) return 0.0;
  int vmin = std::max(std::max(-j1 + j2 + m3, -j1 + m1), 0);
  int vmax = std::min(std::min(j2 + j3 + m1, j3 - j1 + j2), j3 + m3);
  double C = std::sqrt((2.0 * j3 + 1.0) *
      (factd(j3 + j1 - j2) * factd(j3 - j1 + j2) * factd(j1 + j2 - j3) *
       factd(j3 + m3) * factd(j3 - m3)) /
      (factd(j1 + j2 + j3 + 1) * factd(j1 - m1) * factd(j1 + m1) *
       factd(j2 - m2) * factd(j2 + m2)));
  double S = 0.0;
  for (int v = vmin; v <= vmax; ++v) {
    double sgn = ((v + j2 + m2) & 1) ? -1.0 : 1.0;
    S += sgn * (factd(j2 + j3 + m1 - v) * factd(j1 - m1 + v)) /
         (factd(v) * factd(j3 - j1 + j2 - v) * factd(j3 + m3 - v) *
          factd(v + j1 - j2 - m3));
  }
  return C * S;
}

static void qmat(int l, std::complex<double>* q) {
  int d = 2 * l + 1;
  for (int i = 0; i < d * d; ++i) q[i] = {0.0, 0.0};
  const double is2 = 1.0 / std::sqrt(2.0);
  for (int m = -l; m < 0; ++m) {
    q[(l + m) * d + (l - m)] = {is2, 0.0};      // l+|m| column
    q[(l + m) * d + (l + m)] = {0.0, -is2};     // l-|m| column
  }
  q[l * d + l] = {1.0, 0.0};
  for (int m = 1; m <= l; ++m) {
    double s = (m & 1) ? -1.0 : 1.0;
    q[(l + m) * d + (l + m)] = {s * is2, 0.0};
    q[(l + m) * d + (l - m)] = {0.0, s * is2};
  }
  std::complex<double> f(1.0, 0.0), mi(0.0, -1.0);
  for (int i = 0; i < l; ++i) f *= mi;
  for (int i = 0; i < d * d; ++i) q[i] *= f;
}

static void wigner3j_scaled(int l1, int l2, int l3, float* out) {
  int d1 = 2 * l1 + 1, d2 = 2 * l2 + 1, d3 = 2 * l3 + 1;
  std::vector<double> Csu(d1 * d2 * d3);
  for (int m1 = -l1; m1 <= l1; ++m1)
    for (int m2 = -l2; m2 <= l2; ++m2)
      for (int m3 = -l3; m3 <= l3; ++m3)
        Csu[((l1 + m1) * d2 + (l2 + m2)) * d3 + (l3 + m3)] =
            su2_cg(l1, l2, l3, m1, m2, m3);
  std::vector<std::complex<double>> q1(d1 * d1), q2(d2 * d2), q3(d3 * d3);
  qmat(l1, q1.data()); qmat(l2, q2.data()); qmat(l3, q3.data());
  std::vector<double> R(d1 * d2 * d3, 0.0);
  double nrm = 0.0;
  for (int jj = 0; jj < d1; ++jj)
    for (int ll = 0; ll < d2; ++ll)
      for (int mm = 0; mm < d3; ++mm) {
        std::complex<double> s(0.0, 0.0);
        for (int i = 0; i < d1; ++i)
          for (int k = 0; k < d2; ++k)
            for (int nn = 0; nn < d3; ++nn)
              s += q1[i * d1 + jj] * q2[k * d2 + ll] *
                   std::conj(q3[nn * d3 + mm]) * Csu[(i * d2 + k) * d3 + nn];
        R[(jj * d2 + ll) * d3 + mm] = s.real();
        nrm += s.real() * s.real();
      }
  double scale = std::sqrt((double)d3) / std::sqrt(nrm);   // component norm
  for (int x = 0; x < d1 * d2 * d3; ++x) out[x] = (float)(R[x] * scale);
}

struct HPath { int l3, p3, mul1, mul2, d1, d2, a1, a2; std::vector<float> cg; };

static void build_tables(DevTables& T) {
  std::memset(&T, 0, sizeof(T));
  const int muls[3] = {64, 32, 16};
  const int ls[3]   = {0, 1, 2};
  const int ps[3]   = {1, -1, 1};
  const int off[3]  = {0, 64, 160};

  std::vector<HPath> paths;
  for (int i1 = 0; i1 < 3; ++i1)
    for (int i2 = 0; i2 < 3; ++i2)
      for (int l3 = std::abs(ls[i1] - ls[i2]); l3 <= ls[i1] + ls[i2]; ++l3) {
        HPath hp;
        hp.l3 = l3; hp.p3 = ps[i1] * ps[i2];
        hp.mul1 = muls[i1]; hp.mul2 = muls[i2];
        hp.d1 = 2 * ls[i1] + 1; hp.d2 = 2 * ls[i2] + 1;
        hp.a1 = off[i1]; hp.a2 = off[i2];
        hp.cg.resize(hp.d1 * hp.d2 * (2 * l3 + 1));
        wigner3j_scaled(ls[i1], ls[i2], l3, hp.cg.data());
        paths.push_back(std::move(hp));
      }
  const int np = (int)paths.size();  // 19
  std::vector<int> order(np);
  std::iota(order.begin(), order.end(), 0);
  std::stable_sort(order.begin(), order.end(), [&](int a, int b) {
    if (paths[a].l3 != paths[b].l3) return paths[a].l3 < paths[b].l3;
    int ka = -paths[a].p3 * ((paths[a].l3 & 1) ? -1 : 1);
    int kb = -paths[b].p3 * ((paths[b].l3 & 1) ? -1 : 1);
    return ka < kb;
  });
  std::vector<int> ooff(np);
  int acc = 0;
  for (int idx : order) {
    ooff[idx] = acc;
    acc += paths[idx].mul1 * paths[idx].mul2 * (2 * paths[idx].l3 + 1);
  }
  // WMMA paths 0 (0,0)->0, 4 (1,1)->0, 14 (2,2)->0: diagonal CG, grab scale.
  T.s000 = paths[0].cg[0];
  T.s110 = paths[4].cg[0];
  T.s220 = paths[14].cg[0];

  int pi = 0, kpos = 0, nzpos = 0;
  for (int ip = 0; ip < np; ++ip) {
    if (ip == 0 || ip == 4 || ip == 14) continue;
    HPath& hp = paths[ip];
    int d3 = 2 * hp.l3 + 1;
    int* pd = T.pd[pi];
    pd[0] = hp.mul1 - 1;
    pd[1] = (hp.mul1 == 64) ? 6 : ((hp.mul1 == 32) ? 5 : 4);
    pd[2] = hp.d1; pd[3] = hp.d2; pd[4] = d3;
    pd[5] = hp.a1; pd[6] = hp.a2; pd[7] = ooff[ip];
    pd[8] = kpos;  pd[9] = hp.mul1 * hp.mul2;
    for (int k = 0; k < d3; ++k) {
      T.kstarts[kpos + k] = nzpos;
      for (int i = 0; i < hp.d1; ++i)
        for (int j = 0; j < hp.d2; ++j) {
          float v = hp.cg[(i * hp.d2 + j) * d3 + k];
          if (std::fabs(v) > 1e-9f && nzpos < MAXNZ) {
            T.nz_ij[nzpos] = i | (j << 8);
            T.nz_v[nzpos]  = v;
            ++nzpos;
          }
        }
    }
    T.kstarts[kpos + d3] = nzpos;
    kpos += d3 + 1;
    ++pi;
  }
  T.total_nz = nzpos;
}

extern "C" void kernel_launch(void* const* d_in, const int* in_sizes, int n_in,
                              void* d_out, int out_size, void* d_ws, size_t ws_size,
                              hipStream_t stream) {
  const float* in1 = (const float*)d_in[0];
  const float* in2 = (const float*)d_in[1];
  float* out = (float*)d_out;
  const int N = in_sizes[0] / IN_DIM;

  static DevTables h_tables;          // persistent: graph memcpy node references it
  build_tables(h_tables);             // deterministic, identical every call

  hipMemcpyAsync(d_ws, &h_tables, sizeof(DevTables), hipMemcpyHostToDevice, stream);
  ftp_kernel<<<N, 256, 0, stream>>>(in1, in2, out, (const DevTables*)d_ws,
                                    h_tables.total_nz);
}